// RelationLinkModel_73856257622347
// MI455X (gfx1250) — compile-verified
//
#include <hip/hip_runtime.h>

#define H      128
#define KIN    64
#define NSRC   50000
#define NTOT   100000
#define NEDGE  1600000

typedef float v2f __attribute__((ext_vector_type(2)));
typedef float v8f __attribute__((ext_vector_type(8)));

__device__ __forceinline__ v8f wmma4(v2f a, v2f b, v8f c) {
  // D = A(16x4 f32) * B(4x16 f32) + C(16x16 f32)
  return __builtin_amdgcn_wmma_f32_16x16x4_f32(false, a, false, b, (short)0, c,
                                               false, false);
}

// ---------------------------------------------------------------- utilities
__global__ void zero_kernel(float* __restrict__ p, long n4) {
  long i = (long)blockIdx.x * blockDim.x + threadIdx.x;
  long stride = (long)gridDim.x * blockDim.x;
  float4 z = make_float4(0.f, 0.f, 0.f, 0.f);
  for (; i < n4; i += stride) ((float4*)p)[i] = z;
}

__global__ void deg_kernel(const int* __restrict__ dst, float* __restrict__ deg,
                           int E) {
  int i = blockIdx.x * blockDim.x + threadIdx.x;
  if (i < E) unsafeAtomicAdd(&deg[dst[i]], 1.0f);
}

__global__ void invdeg_kernel(const float* __restrict__ deg,
                              float* __restrict__ ideg, int n) {
  int i = blockIdx.x * blockDim.x + threadIdx.x;
  if (i < n) ideg[i] = 1.0f / fmaxf(deg[i], 1.0f);
}

// --------------------------------------------- edge scatter (mean-agg sum)
// one wave per edge; 32 lanes x float4 = 128 floats
__global__ void scatter_kernel(const float* __restrict__ h,
                               const int* __restrict__ src,
                               const int* __restrict__ dst,
                               float* __restrict__ agg, int E) {
  int e = (int)(((long)blockIdx.x * blockDim.x + threadIdx.x) >> 5);
  int lane = threadIdx.x & 31;
  if (e >= E) return;
  int s = src[e];
  int d = dst[e];
  float4 v = ((const float4*)(h + (size_t)s * H))[lane];
  float* ap = agg + (size_t)d * H + lane * 4;
  unsafeAtomicAdd(ap + 0, v.x);
  unsafeAtomicAdd(ap + 1, v.y);
  unsafeAtomicAdd(ap + 2, v.z);
  unsafeAtomicAdd(ap + 3, v.w);
}

// ------------------------------------------- input projection: X @ W.T + b
// X:[M,64] W:[128,64] out:[M,128]; 512 thr = 16 waves;
// wave = 16-row strip x 64-col half (4 tiles) -> acc = 32 VGPRs
__global__ __launch_bounds__(512) void proj_kernel(
    const float* __restrict__ X, const float* __restrict__ W,
    const float* __restrict__ bias, float* __restrict__ out, int M) {
  extern __shared__ float lds[];
  const int WSTR = KIN + 2;  // 66: stride-66 kills LDS bank conflicts
  int tid = threadIdx.x;
  for (int i = tid; i < H * KIN; i += blockDim.x) {
    int r = i >> 6, c = i & (KIN - 1);
    lds[r * WSTR + c] = W[i];
  }
  __syncthreads();

  int wid = tid >> 5, lane = tid & 31;
  int rstrip = wid & 7;   // 8 row strips of 16
  int c0 = (wid >> 3) * 4;  // column half: tiles c0..c0+3
  int row0 = blockIdx.x * 128 + rstrip * 16;
  if (row0 >= M) return;  // wave-uniform: EXEC all-1 where WMMA runs

  int mr = lane & 15;
  int koff = (lane >> 4) << 1;  // lanes 16-31 hold K+2,K+3
  const float* xrow = X + (size_t)(row0 + mr) * KIN + koff;

  const float* wb[4];
#pragma unroll
  for (int c = 0; c < 4; ++c) wb[c] = &lds[((c0 + c) * 16 + mr) * WSTR + koff];

  v8f acc[4] = {};
#pragma unroll
  for (int kk = 0; kk < KIN; kk += 4) {
    v2f a = *(const v2f*)(xrow + kk);
    v2f bf[4];
#pragma unroll
    for (int c = 0; c < 4; ++c) bf[c] = *(const v2f*)(wb[c] + kk);
#pragma unroll
    for (int c = 0; c < 4; ++c) acc[c] = wmma4(a, bf[c], acc[c]);
  }

  int rbase = row0 + ((lane >> 4) << 3);  // lanes>=16 write rows +8
#pragma unroll
  for (int c = 0; c < 4; ++c) {
    int col = (c0 + c) * 16 + mr;
    float bv = bias[col];
#pragma unroll
    for (int r = 0; r < 8; ++r)
      out[(size_t)(rbase + r) * H + col] = acc[c][r] + bv;
  }
}

// ---------- fused SAGE layer: out = (agg*ideg)@Wl.T + bl + h@Wr.T [, relu]
__global__ __launch_bounds__(512) void layer_kernel(
    const float* __restrict__ agg, const float* __restrict__ hin,
    const float* __restrict__ ideg, const float* __restrict__ Wl,
    const float* __restrict__ bias, const float* __restrict__ Wr,
    float* __restrict__ out, int M, int relu) {
  extern __shared__ float lds[];
  const int WSTR = H + 2;  // 130
  float* wl = lds;
  float* wr = lds + H * WSTR;
  int tid = threadIdx.x;
  for (int i = tid; i < H * H; i += blockDim.x) {
    int r = i >> 7, c = i & (H - 1);
    wl[r * WSTR + c] = Wl[i];
    wr[r * WSTR + c] = Wr[i];
  }
  __syncthreads();

  int wid = tid >> 5, lane = tid & 31;
  int rstrip = wid & 7;
  int c0 = (wid >> 3) * 4;
  int row0 = blockIdx.x * 128 + rstrip * 16;
  if (row0 >= M) return;

  int mr = lane & 15;
  int koff = (lane >> 4) << 1;
  size_t rowoff = (size_t)(row0 + mr) * H + koff;
  const float* arow = agg + rowoff;
  const float* hrow = hin + rowoff;
  float scale = ideg[row0 + mr];  // fold mean-normalization into A frag

  const float* wlb[4];
  const float* wrb[4];
#pragma unroll
  for (int c = 0; c < 4; ++c) {
    int o = ((c0 + c) * 16 + mr) * WSTR + koff;
    wlb[c] = wl + o;
    wrb[c] = wr + o;
  }

  v8f acc[4] = {};
#pragma unroll 4
  for (int kk = 0; kk < H; kk += 4) {
    v2f a1 = *(const v2f*)(arow + kk);
    a1 *= scale;
    v2f a2 = *(const v2f*)(hrow + kk);
    v2f b1[4], b2[4];
#pragma unroll
    for (int c = 0; c < 4; ++c) b1[c] = *(const v2f*)(wlb[c] + kk);
#pragma unroll
    for (int c = 0; c < 4; ++c) b2[c] = *(const v2f*)(wrb[c] + kk);
#pragma unroll
    for (int c = 0; c < 4; ++c) {
      acc[c] = wmma4(a1, b1[c], acc[c]);
      acc[c] = wmma4(a2, b2[c], acc[c]);
    }
  }

  int rbase = row0 + ((lane >> 4) << 3);
#pragma unroll
  for (int c = 0; c < 4; ++c) {
    int col = (c0 + c) * 16 + mr;
    float bv = bias[col];
#pragma unroll
    for (int r = 0; r < 8; ++r) {
      float v = acc[c][r] + bv;
      if (relu) v = fmaxf(v, 0.0f);
      out[(size_t)(rbase + r) * H + col] = v;
    }
  }
}

// --------------------------------------------------------------- launcher
extern "C" void kernel_launch(void* const* d_in, const int* in_sizes, int n_in,
                              void* d_out, int out_size, void* d_ws,
                              size_t ws_size, hipStream_t stream) {
  const float* src_x = (const float*)d_in[0];
  const float* tgt_x = (const float*)d_in[1];
  const int* eidx    = (const int*)d_in[2];
  const float* Wsrc  = (const float*)d_in[3];
  const float* bsrc  = (const float*)d_in[4];
  const float* Wtgt  = (const float*)d_in[5];
  const float* btgt  = (const float*)d_in[6];
  const float* Wl    = (const float*)d_in[7];
  const float* bl    = (const float*)d_in[8];
  const float* Wr    = (const float*)d_in[9];

  float* hA   = (float*)d_ws;
  float* hB   = hA + (size_t)NTOT * H;
  float* agg  = hB + (size_t)NTOT * H;
  float* deg  = agg + (size_t)NTOT * H;
  float* ideg = deg + NTOT;

  const int* src_idx = eidx;
  const int* dst_idx = eidx + NEDGE;

  // degree / inverse degree (edge structure fixed; compute once per call)
  zero_kernel<<<512, 256, 0, stream>>>(deg, NTOT / 4);
  deg_kernel<<<(NEDGE + 255) / 256, 256, 0, stream>>>(dst_idx, deg, NEDGE);
  invdeg_kernel<<<(NTOT + 255) / 256, 256, 0, stream>>>(deg, ideg, NTOT);

  // input projections -> hA
  size_t proj_lds = (size_t)H * (KIN + 2) * sizeof(float);  // 33792 B
  proj_kernel<<<(NSRC + 127) / 128, 512, proj_lds, stream>>>(src_x, Wsrc, bsrc,
                                                             hA, NSRC);
  proj_kernel<<<(NSRC + 127) / 128, 512, proj_lds, stream>>>(
      tgt_x, Wtgt, btgt, hA + (size_t)NSRC * H, NSRC);

  // 3 SAGE layers, ping-pong buffers, last writes d_out
  size_t layer_lds = 2 * (size_t)H * (H + 2) * sizeof(float);  // 133120 B
  const float* cur = hA;
  float* nxt = hB;
  for (int i = 0; i < 3; ++i) {
    zero_kernel<<<4096, 256, 0, stream>>>(agg, (long)NTOT * H / 4);
    scatter_kernel<<<(NEDGE + 7) / 8, 256, 0, stream>>>(cur, src_idx, dst_idx,
                                                        agg, NEDGE);
    float* dsto = (i == 2) ? (float*)d_out : nxt;
    layer_kernel<<<(NTOT + 127) / 128, 512, layer_lds, stream>>>(
        agg, cur, ideg, Wl + (size_t)i * H * H, bl + (size_t)i * H,
        Wr + (size_t)i * H * H, dsto, NTOT, (i < 2) ? 1 : 0);
    float* old = (float*)cur;
    cur = dsto;
    nxt = old;
  }
}